// HAN_77335181132168
// MI455X (gfx1250) — compile-verified
//
#include <hip/hip_runtime.h>
#include <math.h>

#define DIMH 128
#define HEADS 8
#define MT 4            // M-tiles (of 16 rows) per block => 64 rows/block
#define ROWS (MT * 16)

typedef __attribute__((ext_vector_type(2))) float v2f;
typedef __attribute__((ext_vector_type(8))) float v8f;

// ---------------- utility ----------------
__device__ __forceinline__ void atomicMaxF(float* addr, float val) {
    int* ai = (int*)addr;
    int old = __float_as_int(*addr);
    while (__int_as_float(old) < val) {
        int assumed = old;
        old = atomicCAS(ai, assumed, __float_as_int(val));
        if (old == assumed) break;
    }
}

__global__ void fill_kernel(float* __restrict__ p, float v, int n) {
    int i = blockIdx.x * blockDim.x + threadIdx.x;
    if (i < n) p[i] = v;
}

__global__ void relu_kernel(float* __restrict__ p, int n) {
    int i = blockIdx.x * blockDim.x + threadIdx.x;
    if (i < n) p[i] = fmaxf(p[i], 0.0f);
}

// ---------------- WMMA GEMM: Y[N,128] = X[N,128] @ W[128,128]^T + bias ----------------
// grid.x = ceil(N/64), block = 256 (8 waves). Wave w owns cols [16w,16w+16) for 4 M-tiles.
// A fragment (16x4 f32): lanes 0-15 -> M=lane, K={k0,k0+1}; lanes 16-31 -> M=lane-16, K={k0+2,k0+3}.
// B fragment (4x16 f32) symmetric; C/D: vgpr g -> M=g (+8 for hi lanes), N=lane&15.
// One B fragment load feeds 4 WMMAs (B reuse across M-tiles).
__global__ void wmma_gemm_bias(const float* __restrict__ X,
                               const float* __restrict__ W,
                               const float* __restrict__ bias,
                               float* __restrict__ Y, int N) {
    __shared__ float xs[ROWS * DIMH];
    const int tile_m = blockIdx.x * ROWS;
    // prefetch all 128 W rows (row tid & 127 per thread) while staging X into LDS
    __builtin_prefetch(W + (size_t)(threadIdx.x & 127) * DIMH, 0, 1);
    for (int e = threadIdx.x; e < ROWS * DIMH; e += blockDim.x) {
        int r = tile_m + (e >> 7);
        r = (r < N) ? r : (N - 1);
        xs[e] = X[(size_t)r * DIMH + (e & 127)];
    }
    __syncthreads();

    const int wave = threadIdx.x >> 5;
    const int lane = threadIdx.x & 31;
    const int l16  = lane & 15;
    const int hi   = (lane >> 4) & 1;
    const int col  = wave * 16 + l16;
    const float* wr = W + (size_t)col * DIMH;   // B[k, col] = W[col, k]

    v8f acc[MT];
#pragma unroll
    for (int t = 0; t < MT; ++t) acc[t] = (v8f){};

    for (int k0 = 0; k0 < DIMH; k0 += 4) {
        const int ka = k0 + (hi << 1);
        v2f b;
        b.x = wr[ka];
        b.y = wr[ka + 1];
#pragma unroll
        for (int t = 0; t < MT; ++t) {
            v2f a;
            a.x = xs[(t * 16 + l16) * DIMH + ka];
            a.y = xs[(t * 16 + l16) * DIMH + ka + 1];
            acc[t] = __builtin_amdgcn_wmma_f32_16x16x4_f32(false, a, false, b,
                                                           (short)0, acc[t], false, false);
        }
    }
    const float bv = bias[col];
#pragma unroll
    for (int t = 0; t < MT; ++t) {
#pragma unroll
        for (int g = 0; g < 8; ++g) {
            const int row = tile_m + t * 16 + g + (hi << 3);
            if (row < N) Y[(size_t)row * DIMH + col] = acc[t][g] + bv;
        }
    }
}

// ---------------- WMMA GEMM + tanh/q reduction for semantic attention ----------------
// score += sum_{rows < N, all 128 cols} tanh(O @ kW^T + kb) * q
__global__ void wmma_gemm_tanh_score(const float* __restrict__ O,
                                     const float* __restrict__ kW,
                                     const float* __restrict__ kb,
                                     const float* __restrict__ q,
                                     float* __restrict__ score_acc, int N) {
    __shared__ float xs[ROWS * DIMH];
    __shared__ float red[8];
    const int tile_m = blockIdx.x * ROWS;
    __builtin_prefetch(kW + (size_t)(threadIdx.x & 127) * DIMH, 0, 1);
    for (int e = threadIdx.x; e < ROWS * DIMH; e += blockDim.x) {
        int r = tile_m + (e >> 7);
        r = (r < N) ? r : (N - 1);
        xs[e] = O[(size_t)r * DIMH + (e & 127)];
    }
    __syncthreads();

    const int wave = threadIdx.x >> 5;
    const int lane = threadIdx.x & 31;
    const int l16  = lane & 15;
    const int hi   = (lane >> 4) & 1;
    const int col  = wave * 16 + l16;
    const float* wr = kW + (size_t)col * DIMH;

    v8f acc[MT];
#pragma unroll
    for (int t = 0; t < MT; ++t) acc[t] = (v8f){};

    for (int k0 = 0; k0 < DIMH; k0 += 4) {
        const int ka = k0 + (hi << 1);
        v2f b;
        b.x = wr[ka];
        b.y = wr[ka + 1];
#pragma unroll
        for (int t = 0; t < MT; ++t) {
            v2f a;
            a.x = xs[(t * 16 + l16) * DIMH + ka];
            a.y = xs[(t * 16 + l16) * DIMH + ka + 1];
            acc[t] = __builtin_amdgcn_wmma_f32_16x16x4_f32(false, a, false, b,
                                                           (short)0, acc[t], false, false);
        }
    }
    const float kbv = kb[col];
    const float qv  = q[col];
    float partial = 0.0f;
#pragma unroll
    for (int t = 0; t < MT; ++t) {
#pragma unroll
        for (int g = 0; g < 8; ++g) {
            const int row = tile_m + t * 16 + g + (hi << 3);
            if (row < N) partial += tanhf(acc[t][g] + kbv) * qv;
        }
    }
    // wave32 reduction
    for (int off = 16; off > 0; off >>= 1)
        partial += __shfl_xor(partial, off, 32);
    if (lane == 0) red[wave] = partial;
    __syncthreads();
    if (threadIdx.x == 0) {
        float s = 0.0f;
#pragma unroll
        for (int w = 0; w < 8; ++w) s += red[w];
        atomicAdd(score_acc, s);
    }
}

// ---------------- per-head attention logits: A[n,h] = sum_d Hf[n,h,d]*att[h,d] ----------------
__global__ void head_score_kernel(const float* __restrict__ Hf,
                                  const float* __restrict__ att,
                                  float* __restrict__ A, int NH) {
    int i = blockIdx.x * blockDim.x + threadIdx.x;
    if (i >= NH) return;
    const int h = i & (HEADS - 1);
    const float* hp = Hf + (size_t)i * 16;   // n*128 + h*16 == i*16
    const float* ap = att + h * 16;
    float s = 0.0f;
#pragma unroll
    for (int d = 0; d < 16; ++d) s += hp[d] * ap[d];
    A[i] = s;
}

// ---------------- edge passes for segment softmax ----------------
__global__ void edge_max_kernel(const float* __restrict__ As, const float* __restrict__ Ad,
                                const int* __restrict__ src, const int* __restrict__ dst,
                                float* __restrict__ M, int E) {
    int e = blockIdx.x * blockDim.x + threadIdx.x;
    if (e >= E) return;
    const int s = src[e], d = dst[e];
#pragma unroll
    for (int h = 0; h < HEADS; ++h) {
        float x = As[s * HEADS + h] + Ad[d * HEADS + h];
        x = (x >= 0.0f) ? x : 0.2f * x;          // leaky_relu 0.2
        atomicMaxF(&M[d * HEADS + h], x);
    }
}

__global__ void edge_expsum_kernel(const float* __restrict__ As, const float* __restrict__ Ad,
                                   const float* __restrict__ M,
                                   const int* __restrict__ src, const int* __restrict__ dst,
                                   float* __restrict__ S, int E) {
    int e = blockIdx.x * blockDim.x + threadIdx.x;
    if (e >= E) return;
    const int s = src[e], d = dst[e];
#pragma unroll
    for (int h = 0; h < HEADS; ++h) {
        float x = As[s * HEADS + h] + Ad[d * HEADS + h];
        x = (x >= 0.0f) ? x : 0.2f * x;
        atomicAdd(&S[d * HEADS + h], expf(x - M[d * HEADS + h]));
    }
}

// one wave per edge; lane covers 4 channels
__global__ void edge_message_kernel(const float* __restrict__ Hsrc,
                                    const float* __restrict__ As, const float* __restrict__ Ad,
                                    const float* __restrict__ M, const float* __restrict__ S,
                                    const int* __restrict__ src, const int* __restrict__ dst,
                                    float* __restrict__ Out, int E) {
    const int gid  = blockIdx.x * blockDim.x + threadIdx.x;
    const int e    = gid >> 5;
    const int lane = gid & 31;
    if (e >= E) return;
    const int s = src[e], d = dst[e];
    const int c0 = lane * 4;
    const int h  = c0 >> 4;
    const float* hp = Hsrc + (size_t)s * DIMH + c0;
    __builtin_prefetch(hp, 0, 1);
    float x = As[s * HEADS + h] + Ad[d * HEADS + h];
    x = (x >= 0.0f) ? x : 0.2f * x;
    const float w = expf(x - M[d * HEADS + h]) / (S[d * HEADS + h] + 1e-16f);
    float* op = Out + (size_t)d * DIMH + c0;
#pragma unroll
    for (int j = 0; j < 4; ++j)
        atomicAdd(&op[j], hp[j] * w);
}

// ---------------- semantic softmax over 2 metapaths ----------------
__global__ void softmax2_kernel(float* __restrict__ sc, float invN) {
    const float s0 = sc[0] * invN, s1 = sc[1] * invN;
    const float m  = fmaxf(s0, s1);
    const float e0 = expf(s0 - m), e1 = expf(s1 - m);
    const float z  = e0 + e1;
    sc[2] = e0 / z;
    sc[3] = e1 / z;
}

// ---------------- output head: [No,2] = fused @ W_out^T + b_out ----------------
__global__ void out_kernel(const float* __restrict__ o1, const float* __restrict__ o2,
                           const float* __restrict__ attn,
                           const float* __restrict__ Wout, const float* __restrict__ bout,
                           float* __restrict__ out, int N) {
    int n = blockIdx.x * blockDim.x + threadIdx.x;
    if (n >= N) return;
    const float a0 = attn[0], a1 = attn[1];
    float acc0 = bout[0], acc1 = bout[1];
    const float* p1 = o1 + (size_t)n * DIMH;
    const float* p2 = o2 + (size_t)n * DIMH;
#pragma unroll 8
    for (int j = 0; j < DIMH; ++j) {
        const float f = a0 * p1[j] + a1 * p2[j];
        acc0 += f * Wout[j];
        acc1 += f * Wout[DIMH + j];
    }
    out[(size_t)n * 2]     = acc0;
    out[(size_t)n * 2 + 1] = acc1;
}

extern "C" void kernel_launch(void* const* d_in, const int* in_sizes, int n_in,
                              void* d_out, int out_size, void* d_ws, size_t ws_size,
                              hipStream_t stream) {
    const float* x_o  = (const float*)d_in[0];
    const float* x_c  = (const float*)d_in[1];
    const float* x_p  = (const float*)d_in[2];
    const float* Wp_o = (const float*)d_in[3];
    const float* bp_o = (const float*)d_in[4];
    const float* Wp_c = (const float*)d_in[5];
    const float* bp_c = (const float*)d_in[6];
    const float* Wp_p = (const float*)d_in[7];
    const float* bp_p = (const float*)d_in[8];
    const float* att_s_pl = (const float*)d_in[9];
    const float* att_d_pl = (const float*)d_in[10];
    const float* att_s_co = (const float*)d_in[11];
    const float* att_d_co = (const float*)d_in[12];
    const float* q    = (const float*)d_in[13];
    const float* kW   = (const float*)d_in[14];
    const float* kb   = (const float*)d_in[15];
    const float* W_out= (const float*)d_in[16];
    const float* b_out= (const float*)d_in[17];
    const int* src_pl = (const int*)d_in[18];
    const int* dst_pl = (const int*)d_in[19];
    const int* src_co = (const int*)d_in[20];
    const int* dst_co = (const int*)d_in[21];

    const int No = in_sizes[0] / DIMH;
    const int Nc = in_sizes[1] / DIMH;
    const int Np = in_sizes[2] / DIMH;
    const int E1 = in_sizes[18];
    const int E2 = in_sizes[20];

    float* ws = (float*)d_ws;
    size_t off = 0;
    float* h_o  = ws + off; off += (size_t)No * DIMH;
    float* h_c  = ws + off; off += (size_t)Nc * DIMH;
    float* h_p  = ws + off; off += (size_t)Np * DIMH;
    float* o1   = ws + off; off += (size_t)No * DIMH;
    float* o2   = ws + off; off += (size_t)No * DIMH;
    float* a_s  = ws + off; off += (size_t)((Nc > Np) ? Nc : Np) * HEADS;
    float* a_d  = ws + off; off += (size_t)No * HEADS;
    float* Mbuf = ws + off; off += (size_t)No * HEADS;
    float* Sbuf = ws + off; off += (size_t)No * HEADS;
    float* sc   = ws + off; off += 4;

    const int T = 256;
    auto blocks  = [](long n, int t) { return (int)((n + t - 1) / t); };
    auto tiles64 = [](int n) { return (n + ROWS - 1) / ROWS; };

    // 1) per-type projections (WMMA GEMMs, 64 rows per block)
    wmma_gemm_bias<<<tiles64(No), T, 0, stream>>>(x_o, Wp_o, bp_o, h_o, No);
    wmma_gemm_bias<<<tiles64(Nc), T, 0, stream>>>(x_c, Wp_c, bp_c, h_c, Nc);
    wmma_gemm_bias<<<tiles64(Np), T, 0, stream>>>(x_p, Wp_p, bp_p, h_p, Np);

    // 2) zero metapath accumulators
    fill_kernel<<<blocks((long)No * DIMH, T), T, 0, stream>>>(o1, 0.0f, No * DIMH);
    fill_kernel<<<blocks((long)No * DIMH, T), T, 0, stream>>>(o2, 0.0f, No * DIMH);

    // 3) GAT edge attention per metapath
    struct { const float* hs; int Ns; const float* as; const float* ad;
             const int* src; const int* dst; int E; float* out; } mp[2] = {
        { h_c, Nc, att_s_pl, att_d_pl, src_pl, dst_pl, E1, o1 },
        { h_p, Np, att_s_co, att_d_co, src_co, dst_co, E2, o2 },
    };
    for (int t = 0; t < 2; ++t) {
        fill_kernel<<<blocks((long)No * HEADS, T), T, 0, stream>>>(Mbuf, -3.0e38f, No * HEADS);
        fill_kernel<<<blocks((long)No * HEADS, T), T, 0, stream>>>(Sbuf, 0.0f, No * HEADS);
        head_score_kernel<<<blocks((long)mp[t].Ns * HEADS, T), T, 0, stream>>>(
            mp[t].hs, mp[t].as, a_s, mp[t].Ns * HEADS);
        head_score_kernel<<<blocks((long)No * HEADS, T), T, 0, stream>>>(
            h_o, mp[t].ad, a_d, No * HEADS);
        edge_max_kernel<<<blocks(mp[t].E, T), T, 0, stream>>>(
            a_s, a_d, mp[t].src, mp[t].dst, Mbuf, mp[t].E);
        edge_expsum_kernel<<<blocks(mp[t].E, T), T, 0, stream>>>(
            a_s, a_d, Mbuf, mp[t].src, mp[t].dst, Sbuf, mp[t].E);
        edge_message_kernel<<<blocks((long)mp[t].E * 32, T), T, 0, stream>>>(
            mp[t].hs, a_s, a_d, Mbuf, Sbuf, mp[t].src, mp[t].dst, mp[t].out, mp[t].E);
        relu_kernel<<<blocks((long)No * DIMH, T), T, 0, stream>>>(mp[t].out, No * DIMH);
    }

    // 4) semantic attention scores (WMMA GEMM + tanh/q reduction)
    fill_kernel<<<1, 4, 0, stream>>>(sc, 0.0f, 2);
    wmma_gemm_tanh_score<<<tiles64(No), T, 0, stream>>>(o1, kW, kb, q, &sc[0], No);
    wmma_gemm_tanh_score<<<tiles64(No), T, 0, stream>>>(o2, kW, kb, q, &sc[1], No);
    softmax2_kernel<<<1, 1, 0, stream>>>(sc, 1.0f / (float)No);

    // 5) fused output head
    out_kernel<<<blocks(No, T), T, 0, stream>>>(o1, o2, sc + 2, W_out, b_out,
                                                (float*)d_out, No);
}